// HausdorffLoss_22299470201559
// MI455X (gfx1250) — compile-verified
//
#include <hip/hip_runtime.h>
#include <hip/hip_bf16.h>
#include <math.h>

// Hausdorff loss for MI455X (gfx1250, wave32).
// seed -> exact 1D EDT along W -> exact 1D EDT along H (+sqrt) ->
// WMMA fp32 (16x16x4) dot-product reduction -> scalar max(l1,l2).

typedef __attribute__((ext_vector_type(2))) float v2f;
typedef __attribute__((ext_vector_type(8))) float v8f;

#define HH 320
#define WW 320
#define BIGF 1.0e8f

#define AS1 __attribute__((address_space(1)))
#define AS3 __attribute__((address_space(3)))

#if defined(__has_builtin)
#if __has_builtin(__builtin_amdgcn_global_load_async_to_lds_b32)
#define HAVE_ASYNC_LDS 1
#endif
#if __has_builtin(__builtin_amdgcn_s_wait_asynccnt)
#define HAVE_WAIT_ASYNC 1
#endif
#endif

// Stage one f32 per lane into LDS. On gfx1250 use the async global->LDS
// path (ASYNCcnt-tracked, no VGPR round trip); otherwise plain copy.
__device__ __forceinline__ void stage_to_lds(float* dst, const float* src) {
#ifdef HAVE_ASYNC_LDS
  __builtin_amdgcn_global_load_async_to_lds_b32(
      (AS1 int*)(void*)src, (AS3 int*)dst, 0, 0);
#else
  *dst = *src;
#endif
}

__device__ __forceinline__ void stage_wait() {
#ifdef HAVE_ASYNC_LDS
#ifdef HAVE_WAIT_ASYNC
  __builtin_amdgcn_s_wait_asynccnt(0);
#else
  asm volatile("s_wait_asynccnt 0x0" ::: "memory");
#endif
#endif
}

__device__ __forceinline__ float sigmf(float x) {
  // 1/(1+e^-x) with v_rcp_f32 instead of the IEEE divide sequence.
  return __builtin_amdgcn_rcpf(1.0f + __expf(-x));
}

// Exact lower envelope: min_j (buf[j] + (i-j)^2), 3 VALU/iter.
// t*t is exact (<2^24) so fmaf matches the reference's square-then-add
// rounding bit-for-bit.
__device__ __forceinline__ float envelope_min(const float* buf, int i, int L) {
  float m = 3.0e38f;
  float t = (float)i;
  for (int j = 0; j < L; j += 4) {
    float t1 = t - 1.0f;
    float t2 = t - 2.0f;
    float t3 = t - 3.0f;
    m = fminf(m, fmaf(t, t, buf[j]));
    m = fminf(m, fmaf(t1, t1, buf[j + 1]));
    m = fminf(m, fmaf(t2, t2, buf[j + 2]));
    m = fminf(m, fmaf(t3, t3, buf[j + 3]));
    t -= 4.0f;
  }
  return m;
}

// ---------------------------------------------------------------- seed maps
__global__ __launch_bounds__(256) void edt_seed(
    const float* __restrict__ logits, const float* __restrict__ target,
    float* __restrict__ gGT, float* __restrict__ gPR, int n) {
  int i = blockIdx.x * blockDim.x + threadIdx.x;
  int stride = gridDim.x * blockDim.x;
  for (; i < n; i += stride) {
    gGT[i] = (target[i] > 0.5f) ? 0.0f : BIGF;
    gPR[i] = (logits[i] > 0.0f) ? 0.0f : BIGF;  // sigmoid(x)>0.5 <=> x>0
  }
}

// ------------------------------------- exact 1D squared EDT along W (rows)
__global__ __launch_bounds__(WW) void edt_pass_w(float* __restrict__ gGT,
                                                 float* __restrict__ gPR) {
  __shared__ float row[WW];
  float* g = (blockIdx.y == 0) ? gGT : gPR;
  long long base = (long long)blockIdx.x * WW;
  int i = threadIdx.x;
  stage_to_lds(&row[i], &g[base + i]);
  stage_wait();
  __syncthreads();
  g[base + i] = envelope_min(row, i, WW);
}

// ---------------------------- exact 1D squared EDT along H (cols) + sqrt
__global__ __launch_bounds__(HH) void edt_pass_h(float* __restrict__ gGT,
                                                 float* __restrict__ gPR) {
  __shared__ float col[HH];
  float* g = (blockIdx.y == 0) ? gGT : gPR;
  int colId = blockIdx.x;  // over B*W columns
  int b = colId / WW;
  int w = colId - b * WW;
  long long base = (long long)b * HH * WW + w;
  int i = threadIdx.x;
  stage_to_lds(&col[i], &g[base + (long long)i * WW]);
  stage_wait();
  __syncthreads();
  g[base + (long long)i * WW] = sqrtf(envelope_min(col, i, HH));
}

// ---------------------------------------------- WMMA fp32 dot reduction
// Per wave: 64-element chunk -> A = 16x4 of p, B = 4x16 of d; with
// A[m][k] = p[base+4m+k], B[k][n] = d[base+4n+k] both use the same
// per-lane address; C[m][m] accumulates the length-4 partial dots.
__global__ __launch_bounds__(256) void wmma_reduce(
    const float* __restrict__ logits, const float* __restrict__ target,
    const float* __restrict__ Dgt, const float* __restrict__ Dpr,
    float* __restrict__ partials, int nchunks, int nblocks) {
  const int lane = threadIdx.x & 31;
  const int waveInBlk = threadIdx.x >> 5;
  const int wavesPerBlk = blockDim.x >> 5;
  const int gw = blockIdx.x * wavesPerBlk + waveInBlk;
  const int nw = nblocks * wavesPerBlk;
  const int off = (lane & 15) * 4 + ((lane >> 4) << 1);

  v8f c1 = {};  // diag accumulates sum p*Dgt
  v8f c2 = {};  // diag accumulates sum t*Dpred
  for (int ch = gw; ch < nchunks; ch += nw) {
    int addr = ch * 64 + off;
    v2f a1;
    a1.x = sigmf(logits[addr]);
    a1.y = sigmf(logits[addr + 1]);
    v2f b1 = *(const v2f*)(Dgt + addr);
    v2f a2 = *(const v2f*)(target + addr);
    v2f b2 = *(const v2f*)(Dpr + addr);
    c1 = __builtin_amdgcn_wmma_f32_16x16x4_f32(false, a1, false, b1,
                                               (short)0, c1, false, false);
    c2 = __builtin_amdgcn_wmma_f32_16x16x4_f32(false, a2, false, b2,
                                               (short)0, c2, false, false);
  }

  // Diagonal: C[v][v] lives in VGPR v at lane v; C[v+8][v+8] at lane v+24.
  float l1 = 0.0f, l2 = 0.0f;
#pragma unroll
  for (int v = 0; v < 8; ++v) {
    bool sel = (lane == v) || (lane == v + 24);
    l1 = sel ? c1[v] : l1;
    l2 = sel ? c2[v] : l2;
  }
  for (int s = 16; s > 0; s >>= 1) {
    l1 += __shfl_down(l1, s, 32);
    l2 += __shfl_down(l2, s, 32);
  }
  __shared__ float s1[8], s2[8];
  if (lane == 0) { s1[waveInBlk] = l1; s2[waveInBlk] = l2; }
  __syncthreads();
  if (threadIdx.x == 0) {
    float t1 = 0.0f, t2 = 0.0f;
    for (int wv = 0; wv < wavesPerBlk; ++wv) { t1 += s1[wv]; t2 += s2[wv]; }
    partials[blockIdx.x] = t1;
    partials[nblocks + blockIdx.x] = t2;
  }
}

// -------------------------------------------------------------- finalize
__global__ __launch_bounds__(256) void finalize(
    const float* __restrict__ partials, int nblocks, float invN,
    float* __restrict__ out) {
  __shared__ float sh1[256], sh2[256];
  float a = 0.0f, b = 0.0f;
  for (int i = threadIdx.x; i < nblocks; i += blockDim.x) {
    a += partials[i];
    b += partials[nblocks + i];
  }
  sh1[threadIdx.x] = a;
  sh2[threadIdx.x] = b;
  __syncthreads();
  for (int s = blockDim.x >> 1; s > 0; s >>= 1) {
    if (threadIdx.x < (unsigned)s) {
      sh1[threadIdx.x] += sh1[threadIdx.x + s];
      sh2[threadIdx.x] += sh2[threadIdx.x + s];
    }
    __syncthreads();
  }
  if (threadIdx.x == 0) out[0] = fmaxf(sh1[0] * invN, sh2[0] * invN);
}

extern "C" void kernel_launch(void* const* d_in, const int* in_sizes, int n_in,
                              void* d_out, int out_size, void* d_ws,
                              size_t ws_size, hipStream_t stream) {
  const float* logits = (const float*)d_in[0];
  const float* target = (const float*)d_in[1];
  float* out = (float*)d_out;

  const int n = in_sizes[0];  // B*1*H*W = 32*320*320
  const int rows = n / WW;    // B*H
  const int cols = n / HH;    // B*W

  float* wsf = (float*)d_ws;
  float* gGT = wsf;                       // n floats -> D_gt
  float* gPR = wsf + (size_t)n;           // n floats -> D_pred
  float* partials = wsf + 2 * (size_t)n;  // 2*NB floats
  const int NB = 512;

  int blocksA = (n + 255) / 256;
  if (blocksA > 4096) blocksA = 4096;
  edt_seed<<<blocksA, 256, 0, stream>>>(logits, target, gGT, gPR, n);
  edt_pass_w<<<dim3(rows, 2), WW, 0, stream>>>(gGT, gPR);
  edt_pass_h<<<dim3(cols, 2), HH, 0, stream>>>(gGT, gPR);

  const int nchunks = n / 64;
  wmma_reduce<<<NB, 256, 0, stream>>>(logits, target, gGT, gPR, partials,
                                      nchunks, NB);
  finalize<<<1, 256, 0, stream>>>(partials, NB, 1.0f / (float)n, out);
}